// WeightedHausdorffDistance_41489384079923
// MI455X (gfx1250) — compile-verified
//
#include <hip/hip_runtime.h>
#include <hip/hip_bf16.h>
#include <math.h>

// ---------------------------------------------------------------------------
// Weighted Hausdorff Distance, single GT point per batch.
//   B=64, H=W=512, P=262144 pixels.  Memory-bound: 64MB of prob_map @23.3TB/s
//   => ~2.8us floor.  Streaming b128 loads, per-lane scalar accumulation,
//   wave collapse via V_WMMA_F32_16X16X4_F32 (B = ones), deterministic
//   two-pass reduction (no float atomics).
// ---------------------------------------------------------------------------

typedef __attribute__((ext_vector_type(2))) float v2f;
typedef __attribute__((ext_vector_type(8))) float v8f;

#define HH 512
#define WW 512
#define NB 64
#define PIXELS (HH * WW)                 // 262144
#define CHUNKS 64                        // blocks per batch
#define THREADS 256                      // 8 waves (wave32)
#define PIX_PER_BLOCK (PIXELS / CHUNKS)  // 4096
#define VEC_PER_THREAD (PIX_PER_BLOCK / THREADS / 4)  // 4 float4 loads / thread

#define MAX_DIST_F 724.07733386f         // sqrt(512^2 + 512^2)
#define EPS_F 1e-6f
#define INV_P_EXP (-1.0f / 9.0f)

// Collapse a per-lane partial to the full wave(32) sum using one
// v_wmma_f32_16x16x4_f32 with B = ones:
//   A: lane L<16 supplies row L at K=0,1; lane L>=16 supplies row L-16 at K=2,3.
//   With a = {partial, 0}: D[m, n] = partial[m] + partial[m+16]  (all n equal).
//   C/D layout: lane n<16 holds rows 0..7 in c[0..7]; lanes 16..31 rows 8..15.
//   => sum(c[0..7]) gives a half-sum per lane; xor-16 shuffle completes it.
__device__ __forceinline__ float wave_sum_wmma(float partial) {
    v2f a;  a.x = partial; a.y = 0.0f;
    v2f b1; b1.x = 1.0f;   b1.y = 1.0f;
    v8f c = {};
    c = __builtin_amdgcn_wmma_f32_16x16x4_f32(
        /*neg_a=*/false, a, /*neg_b=*/false, b1,
        /*c_mod=*/(short)0, c, /*reuse_a=*/false, /*reuse_b=*/false);
    float s = c[0] + c[1] + c[2] + c[3] + c[4] + c[5] + c[6] + c[7];
    s += __shfl_xor(s, 16);
    return s;  // every lane holds the wave sum
}

// Pass 1: each block reduces a 4096-pixel chunk of one batch -> float4 partial.
__global__ void __launch_bounds__(THREADS)
whd_partial_kernel(const float* __restrict__ prob,
                   const float* __restrict__ gt,
                   const float* __restrict__ orig,
                   float4* __restrict__ ws) {
    const int chunk = blockIdx.x;   // 0..CHUNKS-1
    const int b     = blockIdx.y;   // 0..NB-1
    const int tid   = threadIdx.x;

    // Per-batch scalars (norm factor and scaled GT target).
    const float sy = orig[2 * b + 0] * (1.0f / (float)HH);
    const float sx = orig[2 * b + 1] * (1.0f / (float)WW);
    const float ty = gt[2 * b + 0] * sy;
    const float tx = gt[2 * b + 1] * sx;

    const int pixBase = chunk * PIX_PER_BLOCK;
    const float4* pm =
        (const float4*)(prob + (size_t)b * PIXELS + pixBase);

    float sumP = 0.0f, sumPD = 0.0f, sumPW = 0.0f;

#pragma unroll
    for (int j = 0; j < VEC_PER_THREAD; ++j) {
        const int vecIdx = j * THREADS + tid;        // coalesced b128 stream
        const float4 p4 = pm[vecIdx];
        const float pv[4] = {p4.x, p4.y, p4.z, p4.w};
        const int pix0 = pixBase + vecIdx * 4;
#pragma unroll
        for (int k = 0; k < 4; ++k) {
            const int pix = pix0 + k;
            const float y = (float)(pix >> 9);       // W == 512
            const float x = (float)(pix & 511);
            const float dy = fmaf(y, sy, -ty);
            const float dx = fmaf(x, sx, -tx);
            const float d = sqrtf(fmaf(dy, dy, dx * dx));
            const float p = pv[k];
            sumP  += p;
            sumPD += p * d;
            // ((1-p)*MAX_DIST + p*d + eps)^(-9): rcp + 4 mults, no pow.
            const float w = fmaf(p, d, fmaf(-p, MAX_DIST_F, MAX_DIST_F)) + EPS_F;
            const float r  = __builtin_amdgcn_rcpf(w);
            const float r2 = r * r;
            const float r4 = r2 * r2;
            const float r8 = r4 * r4;
            sumPW += r8 * r;
        }
    }

    // Wave-level collapse via WMMA (3 accumulated quantities -> 3 wmma ops).
    const float wP  = wave_sum_wmma(sumP);
    const float wPD = wave_sum_wmma(sumPD);
    const float wPW = wave_sum_wmma(sumPW);

    __shared__ float sP[THREADS / 32], sPD[THREADS / 32], sPW[THREADS / 32];
    const int wave = tid >> 5;
    if ((tid & 31) == 0) {
        sP[wave] = wP; sPD[wave] = wPD; sPW[wave] = wPW;
    }
    __syncthreads();
    if (tid == 0) {
        float aP = 0.0f, aPD = 0.0f, aPW = 0.0f;
#pragma unroll
        for (int wv = 0; wv < THREADS / 32; ++wv) {
            aP += sP[wv]; aPD += sPD[wv]; aPW += sPW[wv];
        }
        ws[(size_t)chunk * NB + b] = make_float4(aP, aPD, aPW, 0.0f);
    }
}

// Pass 2: one 64-thread block; thread b folds its 64 chunk-partials and
// applies the epilogue.  Reads are float4-coalesced across b.
__global__ void __launch_bounds__(NB)
whd_final_kernel(const float4* __restrict__ ws, float* __restrict__ out) {
    const int b = threadIdx.x;
    float aP = 0.0f, aPD = 0.0f, aPW = 0.0f;
#pragma unroll 4
    for (int c = 0; c < CHUNKS; ++c) {
        const float4 v = ws[(size_t)c * NB + b];
        aP += v.x; aPD += v.y; aPW += v.z;
    }
    const float term1 = aPD / (aP + EPS_F);
    const float mean  = aPW * (1.0f / (float)PIXELS);
    const float term2 = powf(mean, INV_P_EXP);
    out[b] = term1 + term2;
}

extern "C" void kernel_launch(void* const* d_in, const int* in_sizes, int n_in,
                              void* d_out, int out_size, void* d_ws, size_t ws_size,
                              hipStream_t stream) {
    const float* prob = (const float*)d_in[0];   // (64, 512, 512) f32
    const float* gt   = (const float*)d_in[1];   // (64, 2) f32
    const float* orig = (const float*)d_in[2];   // (64, 2) f32
    float* out = (float*)d_out;                  // (64,) f32
    float4* ws = (float4*)d_ws;                  // 64*64 float4 = 64 KB

    dim3 grid(CHUNKS, NB, 1);
    whd_partial_kernel<<<grid, THREADS, 0, stream>>>(prob, gt, orig, ws);
    whd_final_kernel<<<1, NB, 0, stream>>>(ws, out);
}